// CombinedModel_24300924961031
// MI455X (gfx1250) — compile-verified
//
#include <hip/hip_runtime.h>
#include <hip/hip_bf16.h>

// ---------------------------------------------------------------------------
// CDNA5 (gfx1250) GCN+MLP combined model.
//  - Dense GEMMs: V_WMMA_F32_16X16X4_F32 (fp32-exact matrix path).
//    Small-K GEMMs stage W in LDS (<=32KB of the 320KB WGP LDS) and sweep all
//    N-tiles per wave so A is read from HBM exactly once.
//  - Graph aggregation: device-built CSR (degree histogram -> exclusive scan
//    -> bucket scatter), then gather-based per-node reduction. This removes
//    ~428M fp32 global atomics (the real roofline) in favor of pure-bandwidth
//    gathers (~1.7GB @ 23.3TB/s).
//  - Pooling: batch is sorted -> per-graph segment reduction, no atomics.
// ---------------------------------------------------------------------------

typedef __attribute__((ext_vector_type(2))) float v2f;
typedef __attribute__((ext_vector_type(8))) float v8f;

__device__ __forceinline__ void atomAddF(float* p, float v) {
  unsafeAtomicAdd(p, v);   // native global_atomic_add_f32
}

// ---------------------------------------------------------------------------
// Direct-load WMMA GEMM (used for the two large embedding GEMMs whose W does
// not fit LDS; W tiles are L2-resident).  C = act(A[M,K] @ W[K,N] + bias)
// One wave per 16x16 C tile, 8 waves/block.
// ---------------------------------------------------------------------------
template <bool BIAS, bool RELU>
__global__ void __launch_bounds__(256)
wmma_gemm_f32(const float* __restrict__ A, int lda,
              const float* __restrict__ W, int ldw,
              const float* __restrict__ bias,
              float* __restrict__ C, int ldc,
              int M, int N, int K) {
  const int tilesN     = N >> 4;
  const int totalTiles = (M >> 4) * tilesN;
  const int tile       = blockIdx.x * 8 + (threadIdx.x >> 5);
  if (tile >= totalTiles) return;              // wave-uniform

  const int tm   = tile / tilesN;
  const int tn   = tile - tm * tilesN;
  const int lane = threadIdx.x & 31;
  const int half = lane >> 4;
  const int l15  = lane & 15;
  const int col  = (tn << 4) + l15;
  const int kb   = half * 2;

  const float* __restrict__ Arow = A + (size_t)((tm << 4) + l15) * lda;
  const float* __restrict__ Wcol = W + col;

  v8f acc = {};
  for (int k = 0; k < K; k += 4) {
    v2f a, b;
    a.x = Arow[k + kb];
    a.y = Arow[k + kb + 1];
    b.x = Wcol[(size_t)(k + kb) * ldw];
    b.y = Wcol[(size_t)(k + kb + 1) * ldw];
    acc = __builtin_amdgcn_wmma_f32_16x16x4_f32(false, a, false, b, (short)0,
                                                acc, false, false);
  }
  const float bv = BIAS ? bias[col] : 0.0f;
#pragma unroll
  for (int j = 0; j < 8; ++j) {
    const int m = (tm << 4) + (half << 3) + j;
    float v = acc[j] + bv;
    if (RELU) v = fmaxf(v, 0.0f);
    C[(size_t)m * ldc + col] = v;
  }
}

// ---------------------------------------------------------------------------
// LDS-staged-W WMMA GEMM for small weight matrices (KK*NN*4 <= 32KB).
// Block of 8 waves loads W into LDS once; each wave owns a 16-row M slab and
// sweeps all NTILES column tiles with NTILES live v8f accumulators, so each
// A element is fetched from global exactly once and B comes from ds_load.
// ---------------------------------------------------------------------------
template <bool BIAS, bool RELU, int KK, int NTILES>
__global__ void __launch_bounds__(256)
wmma_gemm_ldsW(const float* __restrict__ A, int lda,
               const float* __restrict__ W,          // [KK, NN] row-major
               const float* __restrict__ bias,
               float* __restrict__ C, int ldc, int M) {
  constexpr int NN = NTILES * 16;
  __shared__ float sW[KK * NN];
  for (int i = threadIdx.x; i < KK * NN; i += 256) sW[i] = W[i];
  __syncthreads();

  const int tm = blockIdx.x * 8 + (threadIdx.x >> 5);
  if (tm >= (M >> 4)) return;                  // wave-uniform
  const int lane = threadIdx.x & 31;
  const int half = lane >> 4;
  const int l15  = lane & 15;
  const int kb   = half * 2;
  const float* __restrict__ Arow = A + (size_t)((tm << 4) + l15) * lda;

  v8f zero = {};
  v8f acc[NTILES];
#pragma unroll
  for (int t = 0; t < NTILES; ++t) acc[t] = zero;

  for (int k = 0; k < KK; k += 4) {
    v2f a;
    a.x = Arow[k + kb];
    a.y = Arow[k + kb + 1];
#pragma unroll
    for (int t = 0; t < NTILES; ++t) {
      v2f b;
      b.x = sW[(k + kb) * NN + t * 16 + l15];
      b.y = sW[(k + kb + 1) * NN + t * 16 + l15];
      acc[t] = __builtin_amdgcn_wmma_f32_16x16x4_f32(false, a, false, b,
                                                     (short)0, acc[t],
                                                     false, false);
    }
  }
#pragma unroll
  for (int t = 0; t < NTILES; ++t) {
    const int col = t * 16 + l15;
    const float bv = BIAS ? bias[col] : 0.0f;
#pragma unroll
    for (int j = 0; j < 8; ++j) {
      const int m = (tm << 4) + (half << 3) + j;
      float v = acc[t][j] + bv;
      if (RELU) v = fmaxf(v, 0.0f);
      C[(size_t)m * ldc + col] = v;
    }
  }
}

// ---------------------------------------------------------------------------
// Degree / CSR construction
// ---------------------------------------------------------------------------
__global__ void deg_init(float* __restrict__ deg, int n) {
  int i = blockIdx.x * blockDim.x + threadIdx.x;
  if (i < n) deg[i] = 1.0f;                    // self-loop
}
__global__ void deg_count(const int* __restrict__ dst, float* __restrict__ deg,
                          int E) {
  int e = blockIdx.x * blockDim.x + threadIdx.x;
  if (e < E) atomAddF(deg + dst[e], 1.0f);
}
// indeg[i] = deg-1 (real in-edges), deg -> dinv in place
__global__ void finalize_deg(float* __restrict__ deg, int* __restrict__ indeg,
                             int n) {
  int i = blockIdx.x * blockDim.x + threadIdx.x;
  if (i < n) {
    float d = deg[i];
    indeg[i] = (int)(d + 0.5f) - 1;
    deg[i] = rsqrtf(d);                        // d >= 1 always
  }
}

// Exclusive scan of N ints: 512-elem chunks -> chunk sums -> scan sums -> add.
#define SCAN_CHUNK 512
__global__ void scan_chunk(const int* __restrict__ in, int* __restrict__ excl,
                           int* __restrict__ bsum, int n) {
  __shared__ int sh[256];
  const int t = threadIdx.x;
  const int i0 = blockIdx.x * SCAN_CHUNK + 2 * t;
  const int i1 = i0 + 1;
  const int a = (i0 < n) ? in[i0] : 0;
  const int b = (i1 < n) ? in[i1] : 0;
  const int s = a + b;
  sh[t] = s;
  __syncthreads();
  for (int off = 1; off < 256; off <<= 1) {
    int v = (t >= off) ? sh[t - off] : 0;
    __syncthreads();
    sh[t] += v;
    __syncthreads();
  }
  const int ex = sh[t] - s;                    // exclusive prefix of pair sums
  if (i0 < n) excl[i0] = ex;
  if (i1 < n) excl[i1] = ex + a;
  if (t == 255) bsum[blockIdx.x] = sh[255];
}
__global__ void scan_bsum(int* __restrict__ bsum, int nb) {
  __shared__ int sh[1024];
  const int t = threadIdx.x;
  const int v = (t < nb) ? bsum[t] : 0;
  sh[t] = v;
  __syncthreads();
  for (int off = 1; off < 1024; off <<= 1) {
    int u = (t >= off) ? sh[t - off] : 0;
    __syncthreads();
    sh[t] += u;
    __syncthreads();
  }
  if (t < nb) bsum[t] = sh[t] - v;             // exclusive
}
__global__ void scan_add(int* __restrict__ excl, const int* __restrict__ bsum,
                         int n) {
  int i = blockIdx.x * blockDim.x + threadIdx.x;
  if (i < n) excl[i] += bsum[i / SCAN_CHUNK];
}

__global__ void cursor_init(const int* __restrict__ rowstart,
                            int* __restrict__ cursor,
                            int* __restrict__ rowstart_tail, int n, int E) {
  int i = blockIdx.x * blockDim.x + threadIdx.x;
  if (i < n) cursor[i] = rowstart[i];
  if (i == 0) rowstart_tail[0] = E;            // rowstart[N] = E
}
__global__ void edge_bucket(const int* __restrict__ src,
                            const int* __restrict__ dst,
                            int* __restrict__ cursor, int* __restrict__ esrc,
                            int E) {
  int e = blockIdx.x * blockDim.x + threadIdx.x;
  if (e >= E) return;
  int p = atomicAdd(cursor + dst[e], 1);
  esrc[p] = src[e];
}

// ---------------------------------------------------------------------------
// Gather-based GCN aggregation: one wave per node, zero atomics.
// out[i] = act( dinv[i]^2*hw[i] + sum_j dinv[i]*dinv[s_j]*hw[s_j] + bias )
// ---------------------------------------------------------------------------
template <int F, bool RELU>
__global__ void __launch_bounds__(256)
gcn_aggregate(const float* __restrict__ hw, const float* __restrict__ dinv,
              const int* __restrict__ rowstart, const int* __restrict__ esrc,
              const float* __restrict__ bias, float* __restrict__ out, int Nn) {
  const int node = blockIdx.x * 8 + (threadIdx.x >> 5);
  if (node >= Nn) return;
  const int lane = threadIdx.x & 31;
  constexpr int VL = F / 32;                   // floats per lane (2 or 4)
  const int fo = lane * VL;
  const float di = dinv[node];

  float acc[VL];
  {
    const float* hr = hw + (size_t)node * F + fo;
#pragma unroll
    for (int v = 0; v < VL; ++v) acc[v] = di * di * hr[v];
  }
  const int beg = rowstart[node];
  const int end = rowstart[node + 1];
  for (int j = beg; j < end; ++j) {
    const int s = esrc[j];                     // wave-uniform
    const float nrm = di * dinv[s];
    const float* hs = hw + (size_t)s * F + fo;
    if constexpr (VL == 4) {
      const float4 v = *reinterpret_cast<const float4*>(hs);
      acc[0] += nrm * v.x; acc[1] += nrm * v.y;
      acc[2] += nrm * v.z; acc[3] += nrm * v.w;
    } else {
      const float2 v = *reinterpret_cast<const float2*>(hs);
      acc[0] += nrm * v.x; acc[1] += nrm * v.y;
    }
  }
#pragma unroll
  for (int v = 0; v < VL; ++v) {
    float r = acc[v] + bias[fo + v];
    if (RELU) r = fmaxf(r, 0.0f);
    out[(size_t)node * F + fo + v] = r;
  }
}

// ---------------------------------------------------------------------------
// Pooling: batch is sorted -> compute per-graph node ranges, then one wave
// per graph reduces its segment (F = 128).
// ---------------------------------------------------------------------------
__global__ void graph_offsets(const int* __restrict__ batch,
                              int* __restrict__ goff, int Nn, int B) {
  int i = blockIdx.x * blockDim.x + threadIdx.x;
  if (i >= Nn) return;
  int cur  = batch[i];
  int prev = (i == 0) ? -1 : batch[i - 1];
  for (int g = prev + 1; g <= cur; ++g) goff[g] = i;
  if (i == Nn - 1)
    for (int g = cur + 1; g <= B; ++g) goff[g] = Nn;
}
__global__ void __launch_bounds__(256)
pool_mean(const float* __restrict__ h2, const int* __restrict__ goff,
          float* __restrict__ pooled, int B) {
  const int g = blockIdx.x * 8 + (threadIdx.x >> 5);
  if (g >= B) return;
  const int lane = threadIdx.x & 31;
  const int fo = lane * 4;
  const int beg = goff[g], end = goff[g + 1];
  float a0 = 0.f, a1 = 0.f, a2 = 0.f, a3 = 0.f;
  for (int i = beg; i < end; ++i) {
    const float4 v = *reinterpret_cast<const float4*>(h2 + (size_t)i * 128 + fo);
    a0 += v.x; a1 += v.y; a2 += v.z; a3 += v.w;
  }
  const float inv = 1.0f / fmaxf((float)(end - beg), 1.0f);
  float* o = pooled + (size_t)g * 128 + fo;
  o[0] = a0 * inv; o[1] = a1 * inv; o[2] = a2 * inv; o[3] = a3 * inv;
}

// out[g] = dot(c1[g, 0:64], fcf_W) + fcf_b
__global__ void final_fc(const float* __restrict__ c1,
                         const float* __restrict__ w,
                         const float* __restrict__ b,
                         float* __restrict__ out, int B) {
  int g = blockIdx.x * blockDim.x + threadIdx.x;
  if (g >= B) return;
  const float* r = c1 + (size_t)g * 64;
  float acc = b[0];
#pragma unroll
  for (int i = 0; i < 64; ++i) acc = fmaf(r[i], w[i], acc);
  out[g] = acc;
}

// ---------------------------------------------------------------------------
extern "C" void kernel_launch(void* const* d_in, const int* in_sizes, int n_in,
                              void* d_out, int out_size, void* d_ws,
                              size_t ws_size, hipStream_t stream) {
  const float* smiles = (const float*)d_in[0];   // [B,1,768]
  const float* x      = (const float*)d_in[1];   // [N,64]
  const int*   eidx   = (const int*)d_in[2];     // [2,E]
  const int*   batch  = (const int*)d_in[3];     // [N] (sorted)
  const float* emb_W1 = (const float*)d_in[4];   // [768,1024]
  const float* emb_b1 = (const float*)d_in[5];
  const float* emb_W2 = (const float*)d_in[6];   // [1024,64]
  const float* emb_b2 = (const float*)d_in[7];
  const float* c1W    = (const float*)d_in[8];   // [64,64]
  const float* c1b    = (const float*)d_in[9];
  const float* c2W    = (const float*)d_in[10];  // [64,128]
  const float* c2b    = (const float*)d_in[11];
  const float* gfcW   = (const float*)d_in[12];  // [128,64]
  const float* gfcb   = (const float*)d_in[13];
  const float* fc1W   = (const float*)d_in[14];  // [128,64]
  const float* fc1b   = (const float*)d_in[15];
  const float* fcfW   = (const float*)d_in[16];  // [64,1]
  const float* fcfb   = (const float*)d_in[17];
  float* out = (float*)d_out;

  const int B  = in_sizes[0] / 768;   // 4096
  const int Nn = in_sizes[1] / 64;    // 131072
  const int E  = in_sizes[2] / 2;     // 2097152
  const int* src = eidx;
  const int* dst = eidx + E;

  // ---- workspace carve-out (256B-aligned regions) ----
  char* base = (char*)d_ws;
  auto alloc = [&](size_t bytes) -> void* {
    void* p = (void*)base;
    base += (bytes + 255) & ~(size_t)255;
    return p;
  };
  float* dinv     = (float*)alloc((size_t)Nn * 4);          // deg -> dinv
  int*   indeg    = (int*)  alloc((size_t)Nn * 4);
  int*   rowstart = (int*)  alloc(((size_t)Nn + 1) * 4);
  int*   bsum     = (int*)  alloc(1024 * 4);
  int*   cursor   = (int*)  alloc((size_t)Nn * 4);
  int*   esrc     = (int*)  alloc((size_t)E * 4);
  int*   goff     = (int*)  alloc(((size_t)B + 1) * 4);
  float* hw       = (float*)alloc((size_t)Nn * 128 * 4);    // conv GEMM out
  float* out1     = (float*)alloc((size_t)Nn * 64 * 4);
  float* out2     = (float*)alloc((size_t)Nn * 128 * 4);
  float* ehid     = (float*)alloc((size_t)B * 1024 * 4);
  float* comb     = (float*)alloc((size_t)B * 128 * 4);     // concat e|g
  float* pooled   = (float*)alloc((size_t)B * 128 * 4);
  float* c1o      = (float*)alloc((size_t)B * 64 * 4);

  const int T = 256;
  auto nb = [](long long n, int t) { return (int)((n + t - 1) / t); };
  const int nchunks = nb(Nn, SCAN_CHUNK);

  // ---- degrees + CSR build (only ~4M cheap atomics total) ----
  deg_init    <<<nb(Nn, T), T, 0, stream>>>(dinv, Nn);
  deg_count   <<<nb(E, T),  T, 0, stream>>>(dst, dinv, E);
  finalize_deg<<<nb(Nn, T), T, 0, stream>>>(dinv, indeg, Nn);
  scan_chunk  <<<nchunks, 256, 0, stream>>>(indeg, rowstart, bsum, Nn);
  scan_bsum   <<<1, 1024, 0, stream>>>(bsum, nchunks);
  scan_add    <<<nb(Nn, T), T, 0, stream>>>(rowstart, bsum, Nn);
  cursor_init <<<nb(Nn, T), T, 0, stream>>>(rowstart, cursor, rowstart + Nn,
                                            Nn, E);
  edge_bucket <<<nb(E, T), T, 0, stream>>>(src, dst, cursor, esrc, E);
  graph_offsets<<<nb(Nn, T), T, 0, stream>>>(batch, goff, Nn, B);

  // ---- conv1: h1 = relu(aggregate(x @ W1) + b1) ----
  wmma_gemm_ldsW<false, false, 64, 4><<<Nn / 128, 256, 0, stream>>>(
      x, 64, c1W, nullptr, hw, 64, Nn);
  gcn_aggregate<64, true><<<nb(Nn, 8), 256, 0, stream>>>(
      hw, dinv, rowstart, esrc, c1b, out1, Nn);

  // ---- conv2: h2 = aggregate(h1 @ W2) + b2 ----
  wmma_gemm_ldsW<false, false, 64, 8><<<Nn / 128, 256, 0, stream>>>(
      out1, 64, c2W, nullptr, hw, 128, Nn);
  gcn_aggregate<128, false><<<nb(Nn, 8), 256, 0, stream>>>(
      hw, dinv, rowstart, esrc, c2b, out2, Nn);

  // ---- global mean pool (segment reduction, no atomics) ----
  pool_mean<<<nb(B, 8), 256, 0, stream>>>(out2, goff, pooled, B);

  // ---- embedding MLP -> comb[:, 0:64] (large W: direct WMMA path) ----
  {
    int tiles = (B / 16) * (1024 / 16);
    wmma_gemm_f32<true, true><<<nb(tiles, 8), 256, 0, stream>>>(
        smiles, 768, emb_W1, 1024, emb_b1, ehid, 1024, B, 1024, 768);
    tiles = (B / 16) * (64 / 16);
    wmma_gemm_f32<true, false><<<nb(tiles, 8), 256, 0, stream>>>(
        ehid, 1024, emb_W2, 64, emb_b2, comb, 128, B, 64, 1024);
  }
  // ---- gcn_fc -> comb[:, 64:128] ----
  wmma_gemm_ldsW<true, false, 128, 4><<<B / 128, 256, 0, stream>>>(
      pooled, 128, gfcW, gfcb, comb + 64, 128, B);
  // ---- fc1 (no activation in reference) ----
  wmma_gemm_ldsW<true, false, 128, 4><<<B / 128, 256, 0, stream>>>(
      comb, 128, fc1W, fc1b, c1o, 64, B);
  // ---- final 64 -> 1 ----
  final_fc<<<nb(B, T), T, 0, stream>>>(c1o, fcfW, fcfb, out, B);
}